// FPSModule_38826504356625
// MI455X (gfx1250) — compile-verified
//
#include <hip/hip_runtime.h>

// Problem constants (from the reference file)
#define B_DIM   8
#define K_PTS   65536
#define C_DIM   128
#define NSAMP   1024

// FPS kernel tiling
#define NTH          1024                    // threads per block (32 wave32s)
#define CHUNK_PTS    4096                    // points per LDS chunk
#define NCHUNK       (K_PTS / CHUNK_PTS)     // 16
#define PPC          (CHUNK_PTS / NTH)       // 4 points / thread / chunk
#define PPT          (K_PTS / NTH)           // 64 points / thread (dist regs)
#define CHUNK_FLOATS (CHUNK_PTS * 3)         // 12288 floats = 48 KB

// Vector + address-space types matching the builtin's expected signature:
//   __builtin_amdgcn_global_load_async_to_lds_b128(v4i AS1*, v4i AS3*, Ii, Ii)
typedef int v4i __attribute__((ext_vector_type(4)));
typedef __attribute__((address_space(1))) v4i* g1_v4i_ptr;
typedef __attribute__((address_space(3))) v4i* l3_v4i_ptr;

// ---- CDNA5 async global->LDS copy (ASYNCcnt path) -------------------------
__device__ __forceinline__ void async_copy_b128(const void* g, void* l) {
#if __has_builtin(__builtin_amdgcn_global_load_async_to_lds_b128)
  void* gnc = const_cast<void*>(g);  // builtin takes non-const pointer
  __builtin_amdgcn_global_load_async_to_lds_b128(
      (g1_v4i_ptr)gnc, (l3_v4i_ptr)l, /*offset=*/0, /*cpol=*/0);
#else
  unsigned laddr = (unsigned)(unsigned long long)(
      __attribute__((address_space(3))) void*)l;
  asm volatile("global_load_async_to_lds_b128 %0, %1, off"
               :: "v"(laddr), "v"(g) : "memory");
#endif
}

__device__ __forceinline__ void wait_async3() {
#if __has_builtin(__builtin_amdgcn_s_wait_asynccnt)
  __builtin_amdgcn_s_wait_asynccnt(3);
#else
  asm volatile("s_wait_asynccnt 0x3" ::: "memory");
#endif
}

__device__ __forceinline__ void wait_async0() {
#if __has_builtin(__builtin_amdgcn_s_wait_asynccnt)
  __builtin_amdgcn_s_wait_asynccnt(0);
#else
  asm volatile("s_wait_asynccnt 0x0" ::: "memory");
#endif
}

// ---------------------------------------------------------------------------
// One workgroup per batch. dist[] lives in VGPRs; xyz streamed through LDS
// with double-buffered async copies; wave32 shuffle argmax reductions.
// ---------------------------------------------------------------------------
__global__ __launch_bounds__(NTH, 1) void fps_kernel(
    const float* __restrict__ xyz,      // (B, K, 3)
    float*       __restrict__ new_xyz,  // (B, NSAMP, 3)
    int*         __restrict__ inds) {   // (B, NSAMP)
  __shared__ __align__(16) float xbuf[2][CHUNK_FLOATS];  // 96 KB
  __shared__ float redV[32];
  __shared__ int   redI[32];
  __shared__ float bcast[3];

  const int b    = blockIdx.x;
  const int t    = threadIdx.x;
  const int lane = t & 31;
  const int wid  = t >> 5;
  const float* xb = xyz + (size_t)b * (K_PTS * 3);

  // First pick = index 0 (reference semantics)
  if (t == 0) {
    float fx = xb[0], fy = xb[1], fz = xb[2];
    bcast[0] = fx; bcast[1] = fy; bcast[2] = fz;
    inds[b * NSAMP + 0] = 0;
    float* o = new_xyz + (size_t)(b * NSAMP) * 3;
    o[0] = fx; o[1] = fy; o[2] = fz;
  }

  float dist[PPT];
#pragma unroll
  for (int i = 0; i < PPT; ++i) dist[i] = 1e10f;
  __syncthreads();

  for (int n = 1; n < NSAMP; ++n) {
    const float px = bcast[0], py = bcast[1], pz = bcast[2];
    float bv = -1.0f;
    int   bi = 0;

    // Prime the pipeline: async-load chunk 0 into buffer 0.
    {
      const float* g = xb;
      float*       l = xbuf[0];
#pragma unroll
      for (int q = 0; q < 3; ++q)
        async_copy_b128(g + t * 4 + q * 4096, l + t * 4 + q * 4096);
    }

#pragma unroll
    for (int c = 0; c < NCHUNK; ++c) {
      if (c + 1 < NCHUNK) {  // prefetch next chunk into the other buffer
        const float* g = xb + (size_t)(c + 1) * CHUNK_FLOATS;
        float*       l = xbuf[(c + 1) & 1];
#pragma unroll
        for (int q = 0; q < 3; ++q)
          async_copy_b128(g + t * 4 + q * 4096, l + t * 4 + q * 4096);
        wait_async3();     // chunk c's 3 async loads complete (in-order)
      } else {
        wait_async0();
      }
      __syncthreads();     // chunk c visible to all waves

      const float* l = xbuf[c & 1];
#pragma unroll
      for (int r = 0; r < PPC; ++r) {
        const int   lp = r * NTH + t;
        const float x  = l[lp * 3 + 0];
        const float y  = l[lp * 3 + 1];
        const float z  = l[lp * 3 + 2];
        const float dx = x - px, dy = y - py, dz = z - pz;
        const float d  = __fmaf_rn(dx, dx, __fmaf_rn(dy, dy, dz * dz));
        const int   k  = c * PPC + r;
        const float nd = fminf(dist[k], d);
        dist[k] = nd;
        const int j = c * CHUNK_PTS + lp;
        if (nd > bv) { bv = nd; bi = j; }
      }
      __syncthreads();     // done reading this buffer before it is refilled
    }

    // wave32 argmax reduction (tie-break: smaller index, matches jnp.argmax)
#pragma unroll
    for (int off = 16; off > 0; off >>= 1) {
      float ov = __shfl_down(bv, off, 32);
      int   oi = __shfl_down(bi, off, 32);
      if (ov > bv || (ov == bv && oi < bi)) { bv = ov; bi = oi; }
    }
    if (lane == 0) { redV[wid] = bv; redI[wid] = bi; }
    __syncthreads();

    if (wid == 0) {
      bv = redV[lane];
      bi = redI[lane];
#pragma unroll
      for (int off = 16; off > 0; off >>= 1) {
        float ov = __shfl_down(bv, off, 32);
        int   oi = __shfl_down(bi, off, 32);
        if (ov > bv || (ov == bv && oi < bi)) { bv = ov; bi = oi; }
      }
      if (lane == 0) {
        inds[b * NSAMP + n] = bi;
        const float* p = xb + (size_t)bi * 3;   // L2 hit
        float fx = p[0], fy = p[1], fz = p[2];
        bcast[0] = fx; bcast[1] = fy; bcast[2] = fz;
        float* o = new_xyz + ((size_t)b * NSAMP + n) * 3;
        o[0] = fx; o[1] = fy; o[2] = fz;
      }
    }
    __syncthreads();  // broadcast ready; buffers safe for next iteration
  }
}

// ---------------------------------------------------------------------------
// new_features[b][c][n] = features[b][c][inds[b][n]]   (coalesced on n)
// ---------------------------------------------------------------------------
__global__ void gather_feat_kernel(const float* __restrict__ feat,  // (B,C,K)
                                   const int*   __restrict__ inds,  // (B,NSAMP)
                                   float*       __restrict__ out) { // (B,C,NSAMP)
  const int idx  = blockIdx.x * blockDim.x + threadIdx.x;  // B*C*NSAMP
  const int n    = idx & (NSAMP - 1);
  const int bc   = idx >> 10;        // b*C + c
  const int bidx = bc >> 7;          // / C_DIM
  const int ind  = inds[(bidx << 10) + n];
  out[idx] = feat[((size_t)bc << 16) + (unsigned)ind];
}

extern "C" void kernel_launch(void* const* d_in, const int* in_sizes, int n_in,
                              void* d_out, int out_size, void* d_ws, size_t ws_size,
                              hipStream_t stream) {
  (void)in_sizes; (void)n_in; (void)out_size; (void)d_ws; (void)ws_size;

  const float* xyz      = (const float*)d_in[0];  // (B,K,3)
  const float* features = (const float*)d_in[1];  // (B,C,K)

  // Outputs concatenated flat in return order:
  //   new_xyz (B*NSAMP*3 f32) | new_features (B*C*NSAMP f32) | sample_inds (B*NSAMP i32)
  float* out      = (float*)d_out;
  float* new_xyz  = out;
  float* new_feat = out + (size_t)B_DIM * NSAMP * 3;
  int*   inds     = (int*)(out + (size_t)B_DIM * NSAMP * 3 +
                                 (size_t)B_DIM * C_DIM * NSAMP);

  fps_kernel<<<B_DIM, NTH, 0, stream>>>(xyz, new_xyz, inds);

  const int total = B_DIM * C_DIM * NSAMP;  // 1,048,576
  gather_feat_kernel<<<total / 256, 256, 0, stream>>>(features, inds, new_feat);
}